// MoEGate_29583734735605
// MI455X (gfx1250) — compile-verified
//
#include <hip/hip_runtime.h>
#include <hip/hip_bf16.h>

// MoE gate for MI455X (gfx1250), wave32.
//
// logits = hidden[32768,1024] @ weight[8,1024]^T  via V_WMMA_F32_16X16X4_F32:
//   A (16x4)  = weight tile, M = expert (rows 8..15 zero-padded in LDS)
//   B (4x16)  = hidden tile, N = 16 tokens handled by one wave
//   C (16x16) = logits; lane l<16 holds logits[token tok0+l][expert 0..7]
//               in its 8 accumulator VGPRs -> lane-local softmax/top-2.
// Memory-bound problem (~4 FLOP/B): hidden streamed once from HBM,
// weight staged in LDS (64KB incl. zero padding), fp32 kept end to end.

typedef float v2f __attribute__((ext_vector_type(2)));
typedef float v8f __attribute__((ext_vector_type(8)));

constexpr int Hd   = 1024;         // hidden size
constexpr int Ne   = 8;            // experts
constexpr int Sl   = 8192;         // seq len
constexpr int Bb   = 4;            // batch
constexpr int Kk   = 2;            // top-k
constexpr int NTOK = Bb * Sl;      // 32768 tokens
constexpr int NIDX = NTOK * Kk;    // 65536 (size of idx block == weight block)

__global__ __launch_bounds__(256) void moe_gate_kernel(
    const float* __restrict__ hidden, const float* __restrict__ weight,
    float* __restrict__ out, float* __restrict__ part)
{
  __shared__ float w_lds[16 * Hd];    // 64 KB: 8 real expert rows + 8 zero rows
  __shared__ float s_part[8][16];     // per-wave partials: [wave][8 sums | 8 counts]

  const int tid = threadIdx.x;

  // Stage weight into LDS (float4, fully coalesced), zero-pad experts 8..15.
  for (int i = tid; i < (16 * Hd) / 4; i += 256) {
    float4 v = make_float4(0.f, 0.f, 0.f, 0.f);
    if (i < (Ne * Hd) / 4) v = ((const float4*)weight)[i];
    ((float4*)w_lds)[i] = v;
  }
  __syncthreads();

  const int wave = tid >> 5;
  const int lane = tid & 31;
  const int half = lane >> 4;     // 0 -> K pair {k,k+1}, 1 -> {k+2,k+3}
  const int lsub = lane & 15;
  const int tok0 = (blockIdx.x * 8 + wave) * 16;

  // Per-lane operand bases per the documented 32-bit 16x4 A / 4x16 B layouts.
  const float* hbase = hidden + (size_t)(tok0 + lsub) * Hd + 2 * half; // B: token rows
  const float* wbase = w_lds + lsub * Hd + 2 * half;                   // A: expert rows

  v8f c = {0.f, 0.f, 0.f, 0.f, 0.f, 0.f, 0.f, 0.f};
#pragma unroll 8
  for (int k = 0; k < Hd; k += 4) {
    v2f a = *(const v2f*)(wbase + k);   // ds_load_b64 (weight from LDS)
    v2f b = *(const v2f*)(hbase + k);   // global_load_b64 (hidden stream)
    c = __builtin_amdgcn_wmma_f32_16x16x4_f32(false, a, false, b,
                                              (short)0, c, false, false);
  }

  // Lane l<16 now owns logits for token tok0+l, experts 0..7 in c[0..7].
  float sc[8];
  float m = c[0];
#pragma unroll
  for (int e = 1; e < 8; ++e) m = fmaxf(m, c[e]);
  float sum = 0.f;
#pragma unroll
  for (int e = 0; e < 8; ++e) { sc[e] = __expf(c[e] - m); sum += sc[e]; }
  const float inv = 1.0f / sum;
#pragma unroll
  for (int e = 0; e < 8; ++e) sc[e] *= inv;

  // Top-2, strict '>' so ties keep the lowest index (matches jax.lax.top_k).
  float b1 = -1.f, b2 = -1.f;
  int   i1 = 0,    i2 = 0;
#pragma unroll
  for (int e = 0; e < 8; ++e) {
    float s = sc[e];
    if (s > b1)      { b2 = b1; i2 = i1; b1 = s; i1 = e; }
    else if (s > b2) { b2 = s;  i2 = e; }
  }

  const bool active = (lane < 16);
  if (active) {
    const int t = tok0 + lane;
    out[t * 2 + 0] = (float)i1;            // topk_idx (as float; tuple flattened)
    out[t * 2 + 1] = (float)i2;
    out[NIDX + t * 2 + 0] = b1;            // topk_weight
    out[NIDX + t * 2 + 1] = b2;
  }

  // Aux-loss partials: per-expert score sums and top-2 counts over this wave's
  // 16 tokens. shfl-xor tree => deterministic; lanes >=16 contribute zero.
#pragma unroll
  for (int e = 0; e < 8; ++e) {
    float v  = active ? sc[e] : 0.f;
    float cn = active ? ((i1 == e ? 1.f : 0.f) + (i2 == e ? 1.f : 0.f)) : 0.f;
#pragma unroll
    for (int off = 16; off > 0; off >>= 1) {
      v  += __shfl_xor(v,  off, 32);
      cn += __shfl_xor(cn, off, 32);
    }
    if (lane == 0) { s_part[wave][e] = v; s_part[wave][8 + e] = cn; }
  }
  __syncthreads();

  // Fixed-order fold of the 8 waves -> per-block partial in d_ws (deterministic).
  if (tid < 16) {
    float acc = 0.f;
#pragma unroll
    for (int w = 0; w < 8; ++w) acc += s_part[w][tid];
    part[blockIdx.x * 16 + tid] = acc;
  }
}

// Final fold: 256 block-partials -> aux_loss scalar. Each lane owns one (b,e)
// pair (32 = 4 batches x 8 experts); a block never straddles batches since
// 128 tokens/block divides S=8192 (64 blocks per batch).
__global__ void moe_aux_kernel(const float* __restrict__ part,
                               float* __restrict__ out)
{
  const int lane = threadIdx.x;     // 0..31 = b*8 + e
  const int b = lane >> 3;
  const int e = lane & 7;
  float ssum = 0.f, scnt = 0.f;
  for (int blk = b * 64; blk < (b + 1) * 64; ++blk) {
    ssum += part[blk * 16 + e];
    scnt += part[blk * 16 + 8 + e];
  }
  const float p = ssum / (float)Sl;                                   // mean score
  const float f = ((float)Ne / ((float)Kk * (float)Sl)) * scnt / (float)Sl;
  float term = f * p;
#pragma unroll
  for (int off = 16; off > 0; off >>= 1) term += __shfl_xor(term, off, 32);
  if (lane == 0) out[2 * NIDX] = 0.1f * term / (float)Bb;             // ALPHA=0.1
}

extern "C" void kernel_launch(void* const* d_in, const int* in_sizes, int n_in,
                              void* d_out, int out_size, void* d_ws, size_t ws_size,
                              hipStream_t stream) {
  (void)in_sizes; (void)n_in; (void)out_size; (void)ws_size;
  const float* hidden = (const float*)d_in[0];
  const float* weight = (const float*)d_in[1];
  float* out  = (float*)d_out;
  float* part = (float*)d_ws;       // 256 blocks * 16 floats = 16 KB scratch

  // 32768 tokens / (8 waves * 16 tokens per wave) = 256 blocks of 256 threads.
  moe_gate_kernel<<<256, 256, 0, stream>>>(hidden, weight, out, part);
  moe_aux_kernel<<<1, 32, 0, stream>>>(part, out);
}